// DiffusionOrderingNetwork_8495445312023
// MI455X (gfx1250) — compile-verified
//
#include <hip/hip_runtime.h>
#include <hip/hip_bf16.h>
#include <math.h>

typedef __attribute__((ext_vector_type(2))) float v2f;
typedef __attribute__((ext_vector_type(8))) float v8f;

#define TPB 256
#define NEG_SLOPE 0.2f

// ---- monotone unsigned encoding of float for atomicMax-based segment max ----
__device__ __forceinline__ unsigned fenc(float f) {
  unsigned u = __float_as_uint(f);
  return (u & 0x80000000u) ? ~u : (u | 0x80000000u);
}
__device__ __forceinline__ float fdec(unsigned u) {
  u = (u & 0x80000000u) ? (u & 0x7fffffffu) : ~u;
  return __uint_as_float(u);
}

// ---------------------------------------------------------------------------
// Embedding gather: h0[n][c] = emb[x[n]][c]
// ---------------------------------------------------------------------------
__global__ void embed_k(const int* __restrict__ x, const float* __restrict__ emb,
                        float* __restrict__ h0, int N, int D) {
  int i = blockIdx.x * blockDim.x + threadIdx.x;
  if (i >= N * D) return;
  int n = i / D, c = i - n * D;
  h0[i] = emb[x[n] * D + c];
}

// ---------------------------------------------------------------------------
// Full-fp32 WMMA GEMM:  C[n][m] = sum_k A[n][k]*B[k][m] (+ bias[m])
// Requirements: Nrows % 16 == 0, K % 4 == 0 (holds: K in {64,36}, N=50000).
// One wave (32 threads) per 16x16 output tile.
// grid = (ceil(M/16), Nrows/16), block = 32.
//
// VGPR layouts per CDNA5 ISA 7.12.2:
//   A 16x4 f32: lanes 0-15 -> M=lane, v[0]=K0,v[1]=K1; lanes 16-31 -> K2,K3
//   B 4x16 f32: lanes 0-15 -> N=lane, v[0]=K0,v[1]=K1; lanes 16-31 -> K2,K3
//   C/D 16x16 f32: VGPR i, lanes 0-15 -> M=i, N=lane; lanes 16-31 -> M=i+8
// Ragged M handled with clamped loads + value mask so EXEC stays all-ones
// across the WMMA loop (ISA requirement); stores are predicated afterwards.
// ---------------------------------------------------------------------------
__global__ void gemm_wmma_f32(const float* __restrict__ A, const float* __restrict__ B,
                              const float* __restrict__ bias, float* __restrict__ C,
                              int K, int M) {
  const int lane = threadIdx.x & 31;
  const int half = lane >> 4;   // 0: K pair {0,1}, 1: K pair {2,3}
  const int l16  = lane & 15;
  const int row0 = blockIdx.y << 4;
  const int col0 = blockIdx.x << 4;
  const int col  = col0 + l16;
  const int colc = (col < M) ? col : (M - 1);
  const float bmask = (col < M) ? 1.0f : 0.0f;

  v8f acc = {};
  const float* Ap = A + (size_t)(row0 + l16) * K + (half << 1);
  for (int k0 = 0; k0 < K; k0 += 4) {
    v2f a, b;
    a.x = Ap[k0];
    a.y = Ap[k0 + 1];
    const float* Bp = B + (size_t)(k0 + (half << 1)) * M + colc;
    b.x = Bp[0] * bmask;
    b.y = Bp[M] * bmask;
    acc = __builtin_amdgcn_wmma_f32_16x16x4_f32(false, a, false, b, (short)0, acc,
                                                false, false);
  }

  if (col < M) {
    const float bv = bias ? bias[col] : 0.0f;
#pragma unroll
    for (int i = 0; i < 8; ++i) {
      int row = row0 + i + (half << 3);
      C[(size_t)row * M + col] = acc[i] + bv;
    }
  }
}

// ---------------------------------------------------------------------------
// Per-node per-head attention coefficients: a_src[n,h] = <xs[n,h,:], a_s[h,:]>
// i = n*H + h ; xs[(n*H+h)*C + c] = xs + i*C
// ---------------------------------------------------------------------------
__global__ void att_coef_k(const float* __restrict__ xs, const float* __restrict__ a_s,
                           const float* __restrict__ a_d, float* __restrict__ asrc,
                           float* __restrict__ adst, int NH, int H, int C) {
  int i = blockIdx.x * blockDim.x + threadIdx.x;
  if (i >= NH) return;
  int h = i % H;
  const float* xp = xs + (size_t)i * C;
  const float* sp = a_s + h * C;
  const float* dp = a_d + h * C;
  float ss = 0.0f, sd = 0.0f;
  for (int c = 0; c < C; ++c) {
    float v = xp[c];
    ss += v * sp[c];
    sd += v * dp[c];
  }
  asrc[i] = ss;
  adst[i] = sd;
}

// Re-init segment-max (encoded -inf) and segment-sum (0) buffers each call.
__global__ void init_seg_k(unsigned* __restrict__ mEnc, float* __restrict__ ssum, int NH) {
  int i = blockIdx.x * blockDim.x + threadIdx.x;
  if (i >= NH) return;
  mEnc[i] = 0x007FFFFFu;  // fenc(-inf)
  ssum[i] = 0.0f;
}

// Pass 1: e = leaky_relu(a_src[src]+a_dst[dst]); store e; segment max over dst.
__global__ void edge_max_k(const int* __restrict__ ei, int E, int ET,
                           const float* __restrict__ asrc, const float* __restrict__ adst,
                           float* __restrict__ eb, unsigned* __restrict__ mEnc, int H) {
  int e = blockIdx.x * blockDim.x + threadIdx.x;
  if (e >= ET) return;
  int s = (e < E) ? ei[e] : (e - E);
  int d = (e < E) ? ei[E + e] : (e - E);
  for (int h = 0; h < H; ++h) {
    float v = asrc[s * H + h] + adst[d * H + h];
    v = (v > 0.0f) ? v : v * NEG_SLOPE;
    eb[(size_t)e * H + h] = v;
    atomicMax(&mEnc[d * H + h], fenc(v));
  }
}

// Pass 2: ex = exp(e - m[dst]); store ex (in place); segment sum over dst.
__global__ void edge_exp_k(const int* __restrict__ ei, int E, int ET,
                           float* __restrict__ eb, const unsigned* __restrict__ mEnc,
                           float* __restrict__ ssum, int H) {
  int e = blockIdx.x * blockDim.x + threadIdx.x;
  if (e >= ET) return;
  int d = (e < E) ? ei[E + e] : (e - E);
  for (int h = 0; h < H; ++h) {
    float m = fdec(mEnc[d * H + h]);
    float ex = expf(eb[(size_t)e * H + h] - m);
    eb[(size_t)e * H + h] = ex;
    atomicAdd(&ssum[d * H + h], ex);
  }
}

// Pass 3: alpha = ex / (s[dst]+1e-16); out[dst] += xs[src]*alpha
// concat=1: out is N x (H*C);  concat=0: head-mean folded in, out is N x C.
__global__ void edge_scatter_k(const int* __restrict__ ei, int E, int ET,
                               const float* __restrict__ eb, const float* __restrict__ ssum,
                               const float* __restrict__ xs, float* __restrict__ out,
                               int H, int C, int concat) {
  int e = blockIdx.x * blockDim.x + threadIdx.x;
  if (e >= ET) return;
  int s = (e < E) ? ei[e] : (e - E);
  int d = (e < E) ? ei[E + e] : (e - E);
  float alpha[8];
  for (int h = 0; h < H; ++h)
    alpha[h] = eb[(size_t)e * H + h] / (ssum[d * H + h] + 1e-16f);
  const int HC = H * C;
  const float* xp = xs + (size_t)s * HC;
  if (concat) {
    float* op = out + (size_t)d * HC;
    for (int h = 0; h < H; ++h)
      for (int c = 0; c < C; ++c)
        atomicAdd(&op[h * C + c], xp[h * C + c] * alpha[h]);
  } else {
    const float inv = 1.0f / (float)H;
    float* op = out + (size_t)d * C;
    for (int c = 0; c < C; ++c) {
      float acc = 0.0f;
      for (int h = 0; h < H; ++h) acc += xp[h * C + c] * alpha[h];
      atomicAdd(&op[c], acc * inv);
    }
  }
}

__global__ void relu_k(float* __restrict__ p, int n) {
  int i = blockIdx.x * blockDim.x + threadIdx.x;
  if (i < n) p[i] = fmaxf(p[i], 0.0f);
}

// Per-column max and sum(exp(x-max)) over nodes. One block per column.
__global__ void col_reduce_k(const float* __restrict__ X, float* __restrict__ red,
                             int Nr, int M) {
  __shared__ float sm[TPB];
  const int c = blockIdx.x;
  float mx = -INFINITY;
  for (int r = threadIdx.x; r < Nr; r += blockDim.x)
    mx = fmaxf(mx, X[(size_t)r * M + c]);
  sm[threadIdx.x] = mx;
  __syncthreads();
  for (int s = blockDim.x >> 1; s > 0; s >>= 1) {
    if (threadIdx.x < s) sm[threadIdx.x] = fmaxf(sm[threadIdx.x], sm[threadIdx.x + s]);
    __syncthreads();
  }
  mx = sm[0];
  __syncthreads();
  float sum = 0.0f;
  for (int r = threadIdx.x; r < Nr; r += blockDim.x)
    sum += expf(X[(size_t)r * M + c] - mx);
  sm[threadIdx.x] = sum;
  __syncthreads();
  for (int s = blockDim.x >> 1; s > 0; s >>= 1) {
    if (threadIdx.x < s) sm[threadIdx.x] += sm[threadIdx.x + s];
    __syncthreads();
  }
  if (threadIdx.x == 0) {
    red[c] = mx;
    red[M + c] = sm[0];
  }
}

__global__ void softmax_write_k(float* __restrict__ X, const float* __restrict__ red,
                                int NM, int M) {
  int i = blockIdx.x * blockDim.x + threadIdx.x;
  if (i >= NM) return;
  int c = i % M;
  X[i] = expf(X[i] - red[c]) / red[M + c];
}

// ---------------------------------------------------------------------------
// One GATConv layer.
// ---------------------------------------------------------------------------
static void run_layer(const float* h, int Kin, const float* W, const float* a_s,
                      const float* a_d, const float* bias, const float* R, float* out,
                      int C, int concat, int do_relu, const int* ei, int E, int N, int H,
                      float* xs, float* asrc, float* adst, unsigned* mEnc, float* ssum,
                      float* eb, hipStream_t stream) {
  const int Mxs = H * C;
  const int Mout = concat ? (H * C) : C;
  const int ET = E + N;
  dim3 wave(32);
  dim3 gxs((Mxs + 15) / 16, N / 16);
  gemm_wmma_f32<<<gxs, wave, 0, stream>>>(h, W, nullptr, xs, Kin, Mxs);
  dim3 gres((Mout + 15) / 16, N / 16);
  gemm_wmma_f32<<<gres, wave, 0, stream>>>(h, R, bias, out, Kin, Mout);  // residual init
  int NH = N * H;
  att_coef_k<<<(NH + TPB - 1) / TPB, TPB, 0, stream>>>(xs, a_s, a_d, asrc, adst, NH, H, C);
  init_seg_k<<<(NH + TPB - 1) / TPB, TPB, 0, stream>>>(mEnc, ssum, NH);
  edge_max_k<<<(ET + TPB - 1) / TPB, TPB, 0, stream>>>(ei, E, ET, asrc, adst, eb, mEnc, H);
  edge_exp_k<<<(ET + TPB - 1) / TPB, TPB, 0, stream>>>(ei, E, ET, eb, mEnc, ssum, H);
  edge_scatter_k<<<(ET + TPB - 1) / TPB, TPB, 0, stream>>>(ei, E, ET, eb, ssum, xs, out, H,
                                                           C, concat);
  if (do_relu) {
    int n = N * Mout;
    relu_k<<<(n + TPB - 1) / TPB, TPB, 0, stream>>>(out, n);
  }
}

extern "C" void kernel_launch(void* const* d_in, const int* in_sizes, int n_in,
                              void* d_out, int out_size, void* d_ws, size_t ws_size,
                              hipStream_t stream) {
  const int*   x   = (const int*)d_in[0];
  const int*   ei  = (const int*)d_in[1];
  // d_in[2] edge_attr: unused (edge_dim=None in reference)
  const float* emb = (const float*)d_in[3];
  const float* w1  = (const float*)d_in[4];
  const float* as1 = (const float*)d_in[5];
  const float* ad1 = (const float*)d_in[6];
  const float* b1  = (const float*)d_in[7];
  const float* r1  = (const float*)d_in[8];
  const float* w2  = (const float*)d_in[9];
  const float* as2 = (const float*)d_in[10];
  const float* ad2 = (const float*)d_in[11];
  const float* b2  = (const float*)d_in[12];
  const float* r2  = (const float*)d_in[13];
  const float* w3  = (const float*)d_in[14];
  const float* as3 = (const float*)d_in[15];
  const float* ad3 = (const float*)d_in[16];
  const float* b3  = (const float*)d_in[17];
  const float* r3  = (const float*)d_in[18];

  const int N = in_sizes[0];
  const int E = in_sizes[1] / 2;
  const int ET = E + N;
  const int H = 6, D = 64, HID = 36, C1 = HID / H;  // C1 = 6

  // --- workspace carve-out (floats) ---
  float* ws = (float*)d_ws;
  size_t off = 0;
  float* h0 = ws + off;   off += (size_t)N * D;        // embedding / layer-1 input
  float* hA = ws + off;   off += (size_t)N * HID;      // layer-1 output
  float* hB = ws + off;   off += (size_t)N * HID;      // layer-2 output
  float* xs = ws + off;   off += (size_t)N * H * D;    // per-head features (max N*384)
  float* asrc = ws + off; off += (size_t)N * H;
  float* adst = ws + off; off += (size_t)N * H;
  unsigned* mEnc = (unsigned*)(ws + off); off += (size_t)N * H;
  float* ssum = ws + off; off += (size_t)N * H;
  float* eb = ws + off;   off += (size_t)ET * H;       // per-edge logits / exp
  float* red = ws + off;  off += 2 * (size_t)D;        // softmax column max/sum

  // Embedding lookup
  embed_k<<<(N * D + TPB - 1) / TPB, TPB, 0, stream>>>(x, emb, h0, N, D);

  // Layer 1: D -> HID, concat, relu
  run_layer(h0, D, w1, as1, ad1, b1, r1, hA, C1, 1, 1, ei, E, N, H,
            xs, asrc, adst, mEnc, ssum, eb, stream);
  // Layer 2: HID -> HID, concat, relu
  run_layer(hA, HID, w2, as2, ad2, b2, r2, hB, C1, 1, 1, ei, E, N, H,
            xs, asrc, adst, mEnc, ssum, eb, stream);
  // Layer 3: HID -> D, head-mean, no relu; build pre-softmax directly in d_out
  float* outf = (float*)d_out;
  run_layer(hB, HID, w3, as3, ad3, b3, r3, outf, D, 0, 0, ei, E, N, H,
            xs, asrc, adst, mEnc, ssum, eb, stream);

  // softmax over nodes (axis 0), in place on d_out
  col_reduce_k<<<D, TPB, 0, stream>>>(outf, red, N, D);
  softmax_write_k<<<(N * D + TPB - 1) / TPB, TPB, 0, stream>>>(outf, red, N * D, D);
}